// Bilinear_3882650435896
// MI455X (gfx1250) — compile-verified
//
#include <hip/hip_runtime.h>
#include <cstdint>
#include <cmath>

// ---------------------------------------------------------------------------
// Depthwise 2x bilinear upsample (block-diagonal conv_transpose2d, stride 2,
// k=4) for MI455X / gfx1250.
//   in : (N,256,128,128) f32      out : (N,256,258,258) f32
// Memory-bound (~336 MB @ 23.3 TB/s => ~14us floor). Compute is done as two
// banded GEMM passes on the WMMA pipe: out = U_y * X * U_x^T per channel,
// 3x v_wmma_f32_16x16x4_f32 per 16x16 tile, fp32 throughout.
// Input staged global->LDS via async-to-LDS (ASYNCcnt) per CDNA5.
// ---------------------------------------------------------------------------

typedef float v2f __attribute__((ext_vector_type(2)));
typedef float v8f __attribute__((ext_vector_type(8)));

#define CH      256
#define IN_H    128
#define IN_W    128
#define OUT_H   258
#define OUT_W   258
#define XS_STR  131   // col0 = zero (x[-1]), cols1..128 = data, 129/130 = zero
#define TS_STR  274   // 272 live cols (17 tiles * 16) + pad; conflict-free
#define XROWS   18    // input rows 16*band-1 .. 16*band+16
#define TROWS   18

#ifndef USE_ASYNC_LDS
#define USE_ASYNC_LDS 1
#endif

__device__ __forceinline__ v8f wmma_f32_16x16x4(v2f a, v2f b, v8f c) {
#if __has_builtin(__builtin_amdgcn_wmma_f32_16x16x4_f32)
  // (neg_a, A, neg_b, B, c_mod, C, reuse_a, reuse_b) per probe doc
  return __builtin_amdgcn_wmma_f32_16x16x4_f32(false, a, false, b, (short)0, c,
                                               false, false);
#else
#warning "wmma_f32_16x16x4_f32 builtin missing - placeholder fallback (wrong!)"
  c[0] += a[0] * b[0] + a[1] * b[1];
  return c;
#endif
}

extern "C" __global__ void __launch_bounds__(256)
bilinear_up2x_kernel(const float* __restrict__ x, const float* __restrict__ w,
                     float* __restrict__ out) {
  __shared__ float xs[XROWS * XS_STR];  // staged input slab (zero-padded)
  __shared__ float ts[TROWS * TS_STR];  // x-upsampled intermediate

  const int band = blockIdx.x;  // 0..8 : output rows [32*band, 32*band+32)
  const int c    = blockIdx.y;
  const int n    = blockIdx.z;

  const int tid  = threadIdx.x;
  const int lane = tid & 31;   // wave32
  const int wid  = tid >> 5;   // 8 waves
  const int half = lane >> 4;  // 0: lanes 0-15, 1: lanes 16-31
  const int nn   = lane & 15;

  // ---- 1) zero the input staging buffer (halo rows/cols must be 0) --------
  for (int i = tid; i < XROWS * XS_STR; i += 256) xs[i] = 0.0f;
  __syncthreads();

  // ---- 2) async-stage valid input rows into LDS ---------------------------
  const float* xin = x + ((size_t)(n * CH + c) * IN_H) * IN_W;
  const int g0 = 16 * band - 1;
  for (int r = wid; r < XROWS; r += 8) {
    const int g = g0 + r;
    if (g >= 0 && g < IN_H) {
#if USE_ASYNC_LDS
#pragma unroll
      for (int q = 0; q < 4; ++q) {
        uint32_t lds = (uint32_t)(uintptr_t)&xs[r * XS_STR + 1 + q * 32 + lane];
        const float* gp = xin + (size_t)g * IN_W + q * 32 + lane;
        asm volatile("global_load_async_to_lds_b32 %0, %1, off"
                     :: "v"(lds), "v"((uint64_t)(uintptr_t)gp)
                     : "memory");
      }
#else
#pragma unroll
      for (int q = 0; q < 4; ++q)
        xs[r * XS_STR + 1 + q * 32 + lane] =
            xin[(size_t)g * IN_W + q * 32 + lane];
#endif
    }
  }

  // ---- 3) separable taps from the block-diagonal weight -------------------
  // w is (Cin,Cout,4,4); diagonal block base = c*(CH+1)*16. Rank-1 factor:
  // fy[a]*fx[b] = W4[a][1]*W4[1][b]/W4[1][1] == W4[a][b] for separable filt.
  const float* wd = w + (size_t)c * (CH + 1) * 16;
  const float rs = 1.0f / sqrtf(wd[5]);  // 1/sqrt(W4[1][1])
  const float fx0 = wd[4] * rs, fx1 = wd[5] * rs, fx2 = wd[6] * rs,
              fx3 = wd[7] * rs;
  const float fy0 = wd[1] * rs, fy1 = wd[5] * rs, fy2 = wd[9] * rs,
              fy3 = wd[13] * rs;

  // Banded upsample operators in WMMA register layout.
  //   out[2u+p] = f[p]*in[u] + f[p+2]*in[u-1]; window col k <-> in[v0-1+k]
  //   U[k][j]   = (k==(j>>1)+1 ? f[j&1] : 0) + (k==(j>>1) ? f[(j&1)+2] : 0)
  // Bx[s] : B operand (4x16 f32), k = 4s + vgpr + 2*half, n = nn  (pass 1)
  // Ay[s] : A operand (16x4 f32), k = 4s + vgpr + 2*half, m = nn  (pass 2)
  v2f Bx[3], Ay[3];
#pragma unroll
  for (int s = 0; s < 3; ++s) {
#pragma unroll
    for (int r = 0; r < 2; ++r) {
      const int k = 4 * s + r + 2 * half;
      const int u = nn >> 1;
      const float txA = (nn & 1) ? fx1 : fx0, txB = (nn & 1) ? fx3 : fx2;
      const float tyA = (nn & 1) ? fy1 : fy0, tyB = (nn & 1) ? fy3 : fy2;
      Bx[s][r] = (k == u + 1 ? txA : 0.0f) + (k == u ? txB : 0.0f);
      Ay[s][r] = (k == u + 1 ? tyA : 0.0f) + (k == u ? tyB : 0.0f);
    }
  }

#if USE_ASYNC_LDS
#if __has_builtin(__builtin_amdgcn_s_wait_asynccnt)
  __builtin_amdgcn_s_wait_asynccnt(0);
#else
  asm volatile("s_wait_asynccnt 0" ::: "memory");
#endif
#endif
  __syncthreads();

  // ---- 4) pass 1: x-upsample, xs -> ts ------------------------------------
  // 18 t-rows via two 16-row tiles (bases 0 and 2); 17 col-tiles cover 272.
  for (int ti = wid; ti < 34; ti += 8) {
    const int rt    = (ti >= 17) ? 1 : 0;
    const int ctile = ti - rt * 17;
    const int xj0   = ctile << 4;
    const int v0    = ctile << 3;  // input col base for this tile
    const int rbase = rt * 2;
    const float* xrow = &xs[(rbase + nn) * XS_STR];
    v8f acc = {0.f, 0.f, 0.f, 0.f, 0.f, 0.f, 0.f, 0.f};
#pragma unroll
    for (int s = 0; s < 3; ++s) {
      v2f a;
#pragma unroll
      for (int r = 0; r < 2; ++r) {
        const int k = 4 * s + r + 2 * half;
        int cc = v0 + k;            // = (v0 - 1 + k) + 1
        if (cc > 130) cc = 130;     // clamp into zeroed pad (keeps A finite)
        a[r] = xrow[cc];
      }
      acc = wmma_f32_16x16x4(a, Bx[s], acc);
    }
#pragma unroll
    for (int r = 0; r < 8; ++r) {
      const int row = r + 8 * half;  // D row m
      if (rt == 0 || row >= 14)      // tile 2 only contributes rows 16,17
        ts[(rbase + row) * TS_STR + xj0 + nn] = acc[r];
    }
  }
  __syncthreads();

  // ---- 5) pass 2: y-upsample, ts -> out (coalesced 64B row segments) ------
  float* obase = out + ((size_t)(n * CH + c) * OUT_H) * OUT_W;
  for (int ti = wid; ti < 34; ti += 8) {
    const int yt    = (ti >= 17) ? 1 : 0;
    const int ctile = ti - yt * 17;
    const int xj0   = ctile << 4;
    const int yo0   = 32 * band + 16 * yt;
    v8f acc = {0.f, 0.f, 0.f, 0.f, 0.f, 0.f, 0.f, 0.f};
#pragma unroll
    for (int s = 0; s < 3; ++s) {
      v2f b;
#pragma unroll
      for (int r = 0; r < 2; ++r) {
        const int k = 4 * s + r + 2 * half;
        int tr = 8 * yt + k;
        if (tr > 17) tr = 17;  // zero-weight K-pad; clamp keeps B finite
        b[r] = ts[tr * TS_STR + xj0 + nn];
      }
      acc = wmma_f32_16x16x4(Ay[s], b, acc);
    }
    const int xj = xj0 + nn;
    if (xj < OUT_W) {
#pragma unroll
      for (int r = 0; r < 8; ++r) {
        const int yo = yo0 + r + 8 * half;
        if (yo < OUT_H) obase[(size_t)yo * OUT_W + xj] = acc[r];
      }
    }
  }
}

extern "C" void kernel_launch(void* const* d_in, const int* in_sizes, int n_in,
                              void* d_out, int out_size, void* d_ws,
                              size_t ws_size, hipStream_t stream) {
  (void)n_in; (void)out_size; (void)d_ws; (void)ws_size;
  const float* x = (const float*)d_in[0];
  const float* w = (const float*)d_in[1];
  float* out = (float*)d_out;
  const int N = in_sizes[0] / (CH * IN_H * IN_W);  // batch (4)
  dim3 grid(9, CH, N);  // 9 row-bands of 32 output rows (last band = 2 rows)
  bilinear_up2x_kernel<<<grid, 256, 0, stream>>>(x, w, out);
}